// PFNet7_6408091206058
// MI455X (gfx1250) — compile-verified
//
#include <hip/hip_runtime.h>
#include <hip/hip_bf16.h>
#include <math.h>

// ---------------------------------------------------------------------------
// PFNet7 for MI455X (gfx1250, wave32).
// Node MLPs run on V_WMMA_F32_16X16X4_F32 (exact fp32 matrix path):
// each wave owns a 16-node tile; a 16xK @ Kx32 layer = (K/4) x 2 WMMA chain,
// fully unrolled via templates (no branches in the hot GEMM body).
// SGConv scatter uses global_atomic_add_f32.
// ---------------------------------------------------------------------------

typedef float v2f __attribute__((ext_vector_type(2)));
typedef float v8f __attribute__((ext_vector_type(8)));

#define LDW 40          // LDS activation tile row stride (>=38, multiple of 4)
#define NEG_SLOPE 0.01f
#define BN_EPS 1e-5f

struct MlpP {
    const float* W[5];
    const float* b[5];
    const float* bn[4][4];   // [layer][be,g,m,v]  (alphabetical pytree order)
};
struct TailP {
    const float* convW;
    const float* convb;
    MlpP nn2;
    MlpP nn3;
};

__device__ __forceinline__ v8f wmma4(v2f a, v2f b, v8f c) {
    // D = A(16x4,f32) * B(4x16,f32) + C(16x16,f32)
    return __builtin_amdgcn_wmma_f32_16x16x4_f32(
        false, a, false, b, (short)0, c, false, false);
}

// C(16 x NOUT) = act(16 x 4*KB) @ W(KROWS x NOUT).
// act: per-wave LDS tile, row stride LDW, zero-padded beyond valid K.
// ISA f32 A 16x4 layout: lanes 0-15 hold K={0,1} in v0/v1, lanes 16-31 K={2,3}.
// B 4x16 mirrors it: lanes 0-15 rows K=0,1; lanes 16-31 rows K=2,3.
// All bounds are compile-time so the body is branch-free and fully unrolled:
// one burst of global/ds loads, then a back-to-back WMMA chain.
template <int KB, int KROWS, int NOUT>
__device__ __forceinline__ void gemm16(const float* __restrict__ act,
                                       const float* __restrict__ W,
                                       v8f c[2]) {
    constexpr int NT = (NOUT + 15) >> 4;
    const int lane = threadIdx.x & 31;
    const int half = lane >> 4;
    const int l16  = lane & 15;
    const bool jok0 = (NOUT >= 16) || (l16 < NOUT);

    v2f breg[KB][NT];
    v2f areg[KB];

    #pragma unroll
    for (int kb = 0; kb < KB; ++kb) {
        const int k0 = kb * 4 + 2 * half;   // per-half row pair
        #pragma unroll
        for (int nt = 0; nt < NT; ++nt) {
            const int j = nt * 16 + l16;
            const bool jok = (nt == 0) ? jok0 : true;   // NT==2 implies NOUT==32
            v2f b;
            // k0 bounds: max k0 = 4*(KB-1)+2+1 = 4*KB-1; only the last block
            // can exceed KROWS, and the comparison folds at compile time when
            // 4*KB == KROWS.
            b.x = ((4 * KB <= KROWS || k0     < KROWS) && jok) ? W[k0 * NOUT + j]       : 0.0f;
            b.y = ((4 * KB <= KROWS || k0 + 1 < KROWS) && jok) ? W[(k0 + 1) * NOUT + j] : 0.0f;
            breg[kb][nt] = b;
        }
    }
    #pragma unroll
    for (int kb = 0; kb < KB; ++kb) {
        const int k0 = kb * 4 + 2 * half;
        v2f a;
        a.x = act[l16 * LDW + k0];
        a.y = act[l16 * LDW + k0 + 1];
        areg[kb] = a;
    }

    v8f z = {};
    c[0] = z; c[1] = z;
    #pragma unroll
    for (int kb = 0; kb < KB; ++kb) {
        #pragma unroll
        for (int nt = 0; nt < NT; ++nt)
            c[nt] = wmma4(areg[kb], breg[kb][nt], c[nt]);
    }
}

// bias (+ optional BN) (+ optional leaky) and store C tile to an LDS tile.
// C layout: tile nt, vgpr r, lanes 0-15 -> row M=r, col nt*16+l16; lanes 16-31 -> M=r+8.
template <bool BN, bool LEAKY>
__device__ __forceinline__ void post_store_lds(v8f c[2], float* __restrict__ dst,
                                               const float* __restrict__ bias,
                                               const float* __restrict__ g,
                                               const float* __restrict__ be,
                                               const float* __restrict__ m,
                                               const float* __restrict__ v) {
    const int lane = threadIdx.x & 31;
    const int half = lane >> 4;
    const int l16  = lane & 15;
    #pragma unroll
    for (int nt = 0; nt < 2; ++nt) {
        const int j = nt * 16 + l16;
        const float bb = bias[j];
        float scale = 1.0f, shift = 0.0f;
        if (BN) {
            scale = g[j] * rsqrtf(v[j] + BN_EPS);
            shift = be[j] - m[j] * scale;
        }
        #pragma unroll
        for (int r = 0; r < 8; ++r) {
            float val = c[nt][r] + bb;
            if (BN) val = val * scale + shift;
            if (LEAKY) val = (val >= 0.0f) ? val : NEG_SLOPE * val;
            dst[(r + 8 * half) * LDW + j] = val;
        }
    }
}

// --------------------------- nn1: x[N,16] -> h[N,32] ------------------------
__global__ __launch_bounds__(256) void mlp1_kernel(const float* __restrict__ x,
                                                   float* __restrict__ h,
                                                   MlpP p) {
    __shared__ float lds[8 * 2 * 16 * LDW];
    const int wave = threadIdx.x >> 5;
    const int lane = threadIdx.x & 31;
    float* T0 = &lds[wave * 2 * 16 * LDW];
    float* T1 = T0 + 16 * LDW;
    const int base = blockIdx.x * 128 + wave * 16;

    // load 16x16 x tile
    #pragma unroll
    for (int t = 0; t < 8; ++t) {
        int e = t * 32 + lane;
        int r = e >> 4, f = e & 15;
        T0[r * LDW + f] = x[(base + r) * 16 + f];
    }
    __syncthreads();

    v8f c[2];
    // layer 0: 16 -> 32 (+BN+leaky)
    gemm16<4, 16, 32>(T0, p.W[0], c);
    post_store_lds<true, true>(c, T1, p.b[0], p.bn[0][1], p.bn[0][0], p.bn[0][2], p.bn[0][3]);
    __syncthreads();
    // layers 1..3: 32 -> 32 (+BN+leaky), ping-pong T1<->T0
    float* src = T1; float* dst = T0;
    #pragma unroll
    for (int l = 1; l < 4; ++l) {
        gemm16<8, 32, 32>(src, p.W[l], c);
        post_store_lds<true, true>(c, dst, p.b[l],
                                   p.bn[l][1], p.bn[l][0], p.bn[l][2], p.bn[l][3]);
        float* t = src; src = dst; dst = t;
        __syncthreads();
    }
    // final linear 32->32, write h
    gemm16<8, 32, 32>(src, p.W[4], c);
    {
        const int half = lane >> 4, l16 = lane & 15;
        #pragma unroll
        for (int nt = 0; nt < 2; ++nt) {
            const int j = nt * 16 + l16;
            const float bb = p.b[4][j];
            #pragma unroll
            for (int r = 0; r < 8; ++r)
                h[(base + r + 8 * half) * 32 + j] = c[nt][r] + bb;
        }
    }
}

// ------------- tail: agg -> conv(leaky) -> nn2 -> nn3 -> outputs ------------
__global__ __launch_bounds__(256) void tail_kernel(const float* __restrict__ x,
                                                   const float* __restrict__ acc,
                                                   const float* __restrict__ dinv,
                                                   float* __restrict__ out_ids,
                                                   float* __restrict__ out_p4,
                                                   TailP p) {
    __shared__ float lds[8 * 3 * 16 * LDW];
    const int wave = threadIdx.x >> 5;
    const int lane = threadIdx.x & 31;
    float* T0 = &lds[wave * 3 * 16 * LDW];
    float* T1 = T0 + 16 * LDW;
    float* T2 = T1 + 16 * LDW;
    const int base = blockIdx.x * 128 + wave * 16;

    // agg = dinv[i] * acc[i]  (acc already holds self-loop + edge sums of hs)
    #pragma unroll
    for (int t = 0; t < 8; ++t) {
        int e = t * 32 + lane;
        int r = e >> 4, f = e & 15;
        T0[r * LDW + f] = acc[(base + r) * 32 + f] * dinv[base + r];
    }
    // zero T2 pad cols 38,39 (nn3 input K=38 padded to 40)
    { int r = lane >> 1, cc = 38 + (lane & 1); T2[r * LDW + cc] = 0.0f; }
    __syncthreads();

    v8f c[2];
    // SGConv linear + leaky -> h2 in T2 cols 0..31
    gemm16<8, 32, 32>(T0, p.convW, c);
    post_store_lds<false, true>(c, T2, p.convb, nullptr, nullptr, nullptr, nullptr);
    __syncthreads();

    // nn2 hidden layers: T2 -> T0 -> T1 -> T0 -> T1   (T2 preserved)
    float* src = T2; float* dst = T0;
    #pragma unroll
    for (int l = 0; l < 4; ++l) {
        gemm16<8, 32, 32>(src, p.nn2.W[l], c);
        post_store_lds<true, true>(c, dst, p.nn2.b[l],
                                   p.nn2.bn[l][1], p.nn2.bn[l][0], p.nn2.bn[l][2], p.nn2.bn[l][3]);
        src = dst; dst = (dst == T0) ? T1 : T0;
        __syncthreads();
    }
    // nn2 final 32->6: emit cand_ids, also append into T2 cols 32..37
    gemm16<8, 32, 6>(src, p.nn2.W[4], c);
    {
        const int half = lane >> 4, l16 = lane & 15;
        if (l16 < 6) {
            const float bb = p.nn2.b[4][l16];
            #pragma unroll
            for (int r = 0; r < 8; ++r) {
                const float val = c[0][r] + bb;
                const int node = base + r + 8 * half;
                out_ids[node * 6 + l16] = val;
                T2[(r + 8 * half) * LDW + 32 + l16] = val;
            }
        }
    }
    __syncthreads();

    // nn3 layer 0: K=38 (padded 40) from T2
    gemm16<10, 38, 32>(T2, p.nn3.W[0], c);
    post_store_lds<true, true>(c, T0, p.nn3.b[0],
                               p.nn3.bn[0][1], p.nn3.bn[0][0], p.nn3.bn[0][2], p.nn3.bn[0][3]);
    __syncthreads();
    src = T0; dst = T1;
    #pragma unroll
    for (int l = 1; l < 4; ++l) {
        gemm16<8, 32, 32>(src, p.nn3.W[l], c);
        post_store_lds<true, true>(c, dst, p.nn3.b[l],
                                   p.nn3.bn[l][1], p.nn3.bn[l][0], p.nn3.bn[l][2], p.nn3.bn[l][3]);
        float* t = src; src = dst; dst = t;
        __syncthreads();
    }
    // nn3 final 32->4: p4 = x[:,12:16] + out
    gemm16<8, 32, 4>(src, p.nn3.W[4], c);
    {
        const int half = lane >> 4, l16 = lane & 15;
        if (l16 < 4) {
            const float bb = p.nn3.b[4][l16];
            #pragma unroll
            for (int r = 0; r < 8; ++r) {
                const int node = base + r + 8 * half;
                out_p4[node * 4 + l16] = c[0][r] + bb + x[node * 16 + 12 + l16];
            }
        }
    }
}

// ------------------------------- edge kernels -------------------------------
__device__ __forceinline__ void atomAddF(float* p, float v) {
    __hip_atomic_fetch_add(p, v, __ATOMIC_RELAXED, __HIP_MEMORY_SCOPE_AGENT);
}

__global__ void sigmoid_kernel(const float* __restrict__ ea,
                               float* __restrict__ o, int E) {
    int i = blockIdx.x * blockDim.x + threadIdx.x;
    if (i < E) o[i] = 1.0f / (1.0f + expf(-ea[i]));
}
__global__ void fill1_kernel(float* d, int n) {
    int i = blockIdx.x * blockDim.x + threadIdx.x;
    if (i < n) d[i] = 1.0f;                       // self-loop contribution
}
__global__ void count_kernel(const int* __restrict__ col, float* deg, int E) {
    int i = blockIdx.x * blockDim.x + threadIdx.x;
    if (i < E) atomAddF(&deg[col[i]], 1.0f);
}
__global__ void rsqrt_kernel(float* d, int n) {
    int i = blockIdx.x * blockDim.x + threadIdx.x;
    if (i < n) d[i] = rsqrtf(d[i]);
}
// hs = h * dinv[node] (in place), acc initialized with self-loop term hs
__global__ void scale_init_kernel(float* __restrict__ h, float* __restrict__ acc,
                                  const float* __restrict__ dinv, int total) {
    int i = blockIdx.x * blockDim.x + threadIdx.x;
    if (i < total) {
        const float v = h[i] * dinv[i >> 5];
        h[i] = v;
        acc[i] = v;
    }
}
// one thread per (edge, 8-feature chunk): acc[col] += hs[row]
__global__ void scatter_kernel(const int* __restrict__ row, const int* __restrict__ col,
                               const float* __restrict__ hs, float* __restrict__ acc,
                               int E) {
    int t = blockIdx.x * blockDim.x + threadIdx.x;
    if (t >= E * 4) return;
    const int e  = t >> 2;
    const int f0 = (t & 3) * 8;
    const int r = row[e], cdst = col[e];
    const float4* src = reinterpret_cast<const float4*>(hs + (size_t)r * 32 + f0);
    const float4 v0 = src[0], v1 = src[1];
    float* dst = acc + (size_t)cdst * 32 + f0;
    atomAddF(dst + 0, v0.x); atomAddF(dst + 1, v0.y);
    atomAddF(dst + 2, v0.z); atomAddF(dst + 3, v0.w);
    atomAddF(dst + 4, v1.x); atomAddF(dst + 5, v1.y);
    atomAddF(dst + 6, v1.z); atomAddF(dst + 7, v1.w);
}

// -------------------------------- launcher ----------------------------------
// d_in pytree order: x, edge_index, edge_attr, batch, then params flattened
// with sorted dict keys: conv{W,b}, nn1{W[5],b[5],bn[4]x{be,g,m,v}}, nn2{...}, nn3{...}
extern "C" void kernel_launch(void* const* d_in, const int* in_sizes, int n_in,
                              void* d_out, int out_size, void* d_ws, size_t ws_size,
                              hipStream_t stream) {
    const int E = in_sizes[2];          // edge_attr: (E,1)
    const int N = in_sizes[3];          // batch: (N,)

    const float* x  = (const float*)d_in[0];
    const int*   ei = (const int*)d_in[1];
    const float* ea = (const float*)d_in[2];
    const int* row = ei;
    const int* col = ei + E;

    float* out     = (float*)d_out;
    float* out_sig = out;
    float* out_ids = out + (size_t)E;
    float* out_p4  = out + (size_t)E + (size_t)N * 6;

    float* h    = (float*)d_ws;                 // N*32
    float* acc  = h + (size_t)N * 32;           // N*32
    float* dinv = acc + (size_t)N * 32;         // N

    auto F = [&](int i) { return (const float*)d_in[i]; };
    MlpP p1;
    TailP tp;
    tp.convW = F(4); tp.convb = F(5);
    for (int l = 0; l < 5; ++l) { p1.W[l] = F(6 + l);  p1.b[l] = F(11 + l); }
    for (int l = 0; l < 4; ++l)
        for (int q = 0; q < 4; ++q) p1.bn[l][q] = F(16 + 4 * l + q);
    for (int l = 0; l < 5; ++l) { tp.nn2.W[l] = F(32 + l); tp.nn2.b[l] = F(37 + l); }
    for (int l = 0; l < 4; ++l)
        for (int q = 0; q < 4; ++q) tp.nn2.bn[l][q] = F(42 + 4 * l + q);
    for (int l = 0; l < 5; ++l) { tp.nn3.W[l] = F(58 + l); tp.nn3.b[l] = F(63 + l); }
    for (int l = 0; l < 4; ++l)
        for (int q = 0; q < 4; ++q) tp.nn3.bn[l][q] = F(68 + 4 * l + q);

    const int tb = 256;
    sigmoid_kernel<<<(E + tb - 1) / tb, tb, 0, stream>>>(ea, out_sig, E);
    fill1_kernel  <<<(N + tb - 1) / tb, tb, 0, stream>>>(dinv, N);
    count_kernel  <<<(E + tb - 1) / tb, tb, 0, stream>>>(col, dinv, E);
    rsqrt_kernel  <<<(N + tb - 1) / tb, tb, 0, stream>>>(dinv, N);
    mlp1_kernel   <<<N / 128, 256, 0, stream>>>(x, h, p1);
    scale_init_kernel<<<((N * 32) + tb - 1) / tb, tb, 0, stream>>>(h, acc, dinv, N * 32);
    {
        long long st = (long long)E * 4;
        scatter_kernel<<<(unsigned)((st + tb - 1) / tb), tb, 0, stream>>>(row, col, h, acc, E);
    }
    tail_kernel<<<N / 128, 256, 0, stream>>>(x, acc, dinv, out_ids, out_p4, tp);
}